// LocalAttention_19121194402080
// MI455X (gfx1250) — compile-verified
//
#include <hip/hip_runtime.h>
#include <hip/hip_bf16.h>

// LocalAttention for MI455X (gfx1250, wave32).
// Pipeline:
//   K1 qkv_proj  : Q/K/V = x@W + b   (fp32 WMMA 16x16x4, exact precision)
//   K2 attn_core : gather + pos-MLP bias + softmax + weighted V sum -> O (aliases Q)
//   K3 outproj_ln: O@Wo + bo + residual + layernorm (fp32 WMMA + wave reductions)
// Workspace: 3 * N*C * 4B = 153.6 MB (Q[=O], Kf, Vf).

#define NPTS  50000
#define CCH   256
#define NHEAD 16
#define HDIM  16
#define KNBR  16
#define SCALE 0.25f
#define LNEPS 1e-5f

typedef float v2f __attribute__((ext_vector_type(2)));
typedef float v8f __attribute__((ext_vector_type(8)));

// ---------------------------------------------------------------------------
// Kernel 1: fused Q/K/V projection.  grid = 3125 blocks (16 rows each),
// block = 256 threads = 8 waves.  Each wave computes 6 of the 48 16x16 tiles.
// ---------------------------------------------------------------------------
__global__ __launch_bounds__(256)
void qkv_proj_kernel(const float* __restrict__ x,
                     const float* __restrict__ Wq, const float* __restrict__ bq,
                     const float* __restrict__ Wk, const float* __restrict__ bk,
                     const float* __restrict__ Wv, const float* __restrict__ bv,
                     float* __restrict__ Q, float* __restrict__ Kf,
                     float* __restrict__ Vf)
{
    __shared__ float xs[16][CCH + 4];          // +4 pad: conflict-free, 8B aligned
    const int m0 = blockIdx.x * 16;
    const int t  = threadIdx.x;

    for (int i = t; i < 16 * CCH; i += 256) {
        int r = i >> 8, c = i & 255;
        xs[r][c] = x[(size_t)(m0 + r) * CCH + c];
    }
    __syncthreads();

    const int wave  = t >> 5;
    const int lane  = t & 31;
    const int mrow  = lane & 15;               // M row (A) / N col (B,D)
    const int khalf = (lane >> 4) ? 2 : 0;     // K sub-offset per lane half

    for (int tile = wave; tile < 48; tile += 8) {     // uniform per wave
        const int o  = tile >> 4;                      // 0=q 1=k 2=v
        const int n0 = (tile & 15) * 16;               // output column base
        const float* W = (o == 0) ? Wq : (o == 1) ? Wk : Wv;
        const float* b = (o == 0) ? bq : (o == 1) ? bk : bv;
        float*     out = (o == 0) ? Q  : (o == 1) ? Kf : Vf;

        const float bval = b[n0 + mrow];
        v8f acc;
        #pragma unroll
        for (int i = 0; i < 8; ++i) acc[i] = bval;

        #pragma unroll 8
        for (int k = 0; k < CCH; k += 4) {
            v2f a, bb;
            a[0]  = xs[mrow][k + khalf];
            a[1]  = xs[mrow][k + khalf + 1];
            bb[0] = W[(size_t)(k + khalf) * CCH + n0 + mrow];
            bb[1] = W[(size_t)(k + khalf + 1) * CCH + n0 + mrow];
            acc = __builtin_amdgcn_wmma_f32_16x16x4_f32(
                      false, a, false, bb, (short)0, acc, false, false);
        }

        #pragma unroll
        for (int r = 0; r < 8; ++r) {
            const int row = (lane < 16) ? r : (r + 8);
            out[(size_t)(m0 + row) * CCH + n0 + mrow] = acc[r];
        }
    }
}

// ---------------------------------------------------------------------------
// Kernel 2: per-point attention core.  grid = 50000, block = 256.
// Thread t = (h,k) with h = t>>4, k = t&15 for the logit/bias phase,
// then t = (h,d) for the output phase.  O aliases Q (row-private).
// ---------------------------------------------------------------------------
__global__ __launch_bounds__(256)
void attn_core_kernel(const float* Qb, const float* __restrict__ Kf,
                      const float* __restrict__ Vf,
                      const long long* __restrict__ idx,
                      const float* __restrict__ pos,
                      const float* __restrict__ W1, const float* __restrict__ b1,
                      const float* __restrict__ W2, const float* __restrict__ b2,
                      float* Ob)
{
    __shared__ float q_s[CCH];
    __shared__ int   jn_s[KNBR];
    __shared__ float rel_s[KNBR][3];
    __shared__ float hid_s[KNBR][64];
    __shared__ float a_s[NHEAD][KNBR];

    const int n = blockIdx.x;
    const int t = threadIdx.x;

    q_s[t] = Qb[(size_t)n * CCH + t];
    if (t < KNBR) jn_s[t] = (int)idx[(size_t)n * KNBR + t];
    __syncthreads();

    if (t < KNBR * 3) {
        const int k = t / 3, c = t - 3 * k;
        rel_s[k][c] = pos[(size_t)jn_s[k] * 3 + c] - pos[(size_t)n * 3 + c];
    }
    __syncthreads();

    // pos-MLP hidden layer: thread computes 4 of the 64 hidden units of one k
    {
        const int k  = t >> 4;
        const int u0 = (t & 15) * 4;
        const float r0 = rel_s[k][0], r1 = rel_s[k][1], r2 = rel_s[k][2];
        #pragma unroll
        for (int uu = 0; uu < 4; ++uu) {
            const int u = u0 + uu;
            const float hsum = r0 * W1[u] + r1 * W1[64 + u] + r2 * W1[128 + u] + b1[u];
            hid_s[k][u] = 0.5f * hsum * (1.0f + erff(hsum * 0.70710678118654752f));
        }
    }
    __syncthreads();

    // logits + bias + softmax over the 16-neighbor group (half-wave shuffles)
    {
        const int h = t >> 4, k = t & 15;
        const float* krow = Kf + (size_t)jn_s[k] * CCH + h * HDIM;
        float dot = 0.0f;
        #pragma unroll
        for (int d = 0; d < HDIM; ++d) dot += q_s[h * HDIM + d] * krow[d];

        float bias = b2[h];
        #pragma unroll 8
        for (int u = 0; u < 64; ++u) bias += hid_s[k][u] * W2[u * NHEAD + h];

        float lg = dot * SCALE + bias;
        float mx = lg;
        #pragma unroll
        for (int m = 8; m >= 1; m >>= 1) mx = fmaxf(mx, __shfl_xor(mx, m, 32));
        const float e = __expf(lg - mx);
        float s = e;
        #pragma unroll
        for (int m = 8; m >= 1; m >>= 1) s += __shfl_xor(s, m, 32);
        a_s[h][k] = e / s;
    }
    __syncthreads();

    // weighted V sum: thread (h,d); per-k loads are fully coalesced (1KB/row)
    {
        const int h = t >> 4, d = t & 15;
        float o = 0.0f;
        #pragma unroll
        for (int k = 0; k < KNBR; ++k)
            o += a_s[h][k] * Vf[(size_t)jn_s[k] * CCH + h * HDIM + d];
        Ob[(size_t)n * CCH + t] = o;   // t == h*16 + d; O aliases Q, row-private
    }
}

// ---------------------------------------------------------------------------
// Kernel 3: output projection + residual + layernorm.
// grid = 3125, block = 256 (8 waves, 2 column-tiles each), WMMA f32 16x16x4.
// ---------------------------------------------------------------------------
__global__ __launch_bounds__(256)
void outproj_ln_kernel(const float* __restrict__ O, const float* __restrict__ x,
                       const float* __restrict__ Wo, const float* __restrict__ bo,
                       const float* __restrict__ gamma, const float* __restrict__ beta,
                       float* __restrict__ y)
{
    __shared__ float os[16][CCH + 4];
    __shared__ float ys[16][CCH + 4];
    const int m0 = blockIdx.x * 16;
    const int t  = threadIdx.x;

    for (int i = t; i < 16 * CCH; i += 256) {
        int r = i >> 8, c = i & 255;
        os[r][c] = O[(size_t)(m0 + r) * CCH + c];
    }
    __syncthreads();

    const int wave  = t >> 5;
    const int lane  = t & 31;
    const int mrow  = lane & 15;
    const int khalf = (lane >> 4) ? 2 : 0;

    #pragma unroll
    for (int i2 = 0; i2 < 2; ++i2) {
        const int n0 = (wave * 2 + i2) * 16;
        const float bval = bo[n0 + mrow];
        v8f acc;
        #pragma unroll
        for (int i = 0; i < 8; ++i) acc[i] = bval;

        #pragma unroll 8
        for (int k = 0; k < CCH; k += 4) {
            v2f a, bb;
            a[0]  = os[mrow][k + khalf];
            a[1]  = os[mrow][k + khalf + 1];
            bb[0] = Wo[(size_t)(k + khalf) * CCH + n0 + mrow];
            bb[1] = Wo[(size_t)(k + khalf + 1) * CCH + n0 + mrow];
            acc = __builtin_amdgcn_wmma_f32_16x16x4_f32(
                      false, a, false, bb, (short)0, acc, false, false);
        }

        #pragma unroll
        for (int r = 0; r < 8; ++r) {
            const int row = (lane < 16) ? r : (r + 8);
            ys[row][n0 + mrow] = acc[r] + x[(size_t)(m0 + row) * CCH + n0 + mrow];
        }
    }
    __syncthreads();

    // layernorm: wave w handles rows 2w and 2w+1, wave32 shuffle reduction
    #pragma unroll
    for (int rr = 0; rr < 2; ++rr) {
        const int row = wave * 2 + rr;
        float s = 0.0f, ss = 0.0f;
        #pragma unroll
        for (int c = lane; c < CCH; c += 32) {
            const float v = ys[row][c];
            s += v; ss += v * v;
        }
        #pragma unroll
        for (int m = 16; m >= 1; m >>= 1) {
            s  += __shfl_xor(s,  m, 32);
            ss += __shfl_xor(ss, m, 32);
        }
        const float mu   = s * (1.0f / CCH);
        const float var  = ss * (1.0f / CCH) - mu * mu;
        const float rstd = rsqrtf(var + LNEPS);
        #pragma unroll
        for (int c = lane; c < CCH; c += 32)
            y[(size_t)(m0 + row) * CCH + c] =
                (ys[row][c] - mu) * rstd * gamma[c] + beta[c];
    }
}

// ---------------------------------------------------------------------------
extern "C" void kernel_launch(void* const* d_in, const int* in_sizes, int n_in,
                              void* d_out, int out_size, void* d_ws, size_t ws_size,
                              hipStream_t stream)
{
    const float*     x     = (const float*)d_in[0];
    const float*     pos   = (const float*)d_in[1];
    const long long* idx   = (const long long*)d_in[2];
    const float*     Wq    = (const float*)d_in[3];
    const float*     bq    = (const float*)d_in[4];
    const float*     Wk    = (const float*)d_in[5];
    const float*     bk    = (const float*)d_in[6];
    const float*     Wv    = (const float*)d_in[7];
    const float*     bv    = (const float*)d_in[8];
    const float*     Wo    = (const float*)d_in[9];
    const float*     bo    = (const float*)d_in[10];
    const float*     W1    = (const float*)d_in[11];
    const float*     b1    = (const float*)d_in[12];
    const float*     W2    = (const float*)d_in[13];
    const float*     b2    = (const float*)d_in[14];
    const float*     gamma = (const float*)d_in[15];
    const float*     beta  = (const float*)d_in[16];
    float*           out   = (float*)d_out;

    const size_t rowElems = (size_t)NPTS * CCH;       // 12.8M floats
    float* Q  = (float*)d_ws;                          // also O after attn_core
    float* Kf = Q  + rowElems;
    float* Vf = Kf + rowElems;                         // total 153.6 MB

    const int mtiles = NPTS / 16;                      // 3125 exactly

    qkv_proj_kernel<<<mtiles, 256, 0, stream>>>(x, Wq, bq, Wk, bk, Wv, bv,
                                                Q, Kf, Vf);
    attn_core_kernel<<<NPTS, 256, 0, stream>>>(Q, Kf, Vf, idx, pos,
                                               W1, b1, W2, b2, Q /* O alias */);
    outproj_ln_kernel<<<mtiles, 256, 0, stream>>>(Q, x, Wo, bo, gamma, beta, out);
}